// DenseCRF_85504208928868
// MI455X (gfx1250) — compile-verified
//
#include <hip/hip_runtime.h>
#include <math.h>

#define CNUM 21
#define HH 512
#define WW 512
#define BB 8
#define TW 32
#define TH 16
#define NPIX (TW*TH)      // 512 pixels per tile
#define SCR_STRIDE 524    // padded so k+2 row (reads) and m+8 row (writes) hit disjoint bank groups

typedef __attribute__((ext_vector_type(2))) float v2f;
typedef __attribute__((ext_vector_type(8))) float v8f;

// ---------------------------------------------------------------------------
// Kernel 1: edge = exp(-|sobel(gray)|), norm = avgpool3(edge) + 1e-6
// Tile 32x16 with halo-2 gray staging in LDS.
// ---------------------------------------------------------------------------
__global__ __launch_bounds__(256) void crf_edge_norm(
    const float* __restrict__ img, float* __restrict__ edge, float* __restrict__ norm)
{
  __shared__ float gt[20][37];   // gray, halo 2
  __shared__ float ed[18][35];   // edge, halo 1

  const int tid = threadIdx.x;
  const int x0 = blockIdx.x * TW, y0 = blockIdx.y * TH, b = blockIdx.z;
  const size_t HW = (size_t)HH * WW;
  const float* rC = img + ((size_t)b * 3 + 0) * HW;
  const float* gC = img + ((size_t)b * 3 + 1) * HW;
  const float* bC = img + ((size_t)b * 3 + 2) * HW;

  for (int idx = tid; idx < 20 * 36; idx += 256) {
    int j = idx / 36, i = idx % 36;
    int ay = y0 - 2 + j, ax = x0 - 2 + i;
    float v = 0.f;
    if (ay >= 0 && ay < HH && ax >= 0 && ax < WW) {
      size_t o = (size_t)ay * WW + ax;
      v = 0.299f * rC[o] + 0.587f * gC[o] + 0.114f * bC[o];
    }
    gt[j][i] = v;
  }
  __syncthreads();

  for (int idx = tid; idx < 18 * 34; idx += 256) {
    int j = idx / 34, i = idx % 34;
    float gx = -gt[j][i]     + gt[j][i+2]
               - 2.f*gt[j+1][i] + 2.f*gt[j+1][i+2]
               - gt[j+2][i]   + gt[j+2][i+2];
    float gy = -gt[j][i] - 2.f*gt[j][i+1] - gt[j][i+2]
               + gt[j+2][i] + 2.f*gt[j+2][i+1] + gt[j+2][i+2];
    float mag = sqrtf(gx*gx + gy*gy + 1e-6f);
    int ay = y0 - 1 + j, ax = x0 - 1 + i;
    ed[j][i] = (ay >= 0 && ay < HH && ax >= 0 && ax < WW) ? __expf(-mag) : 0.f;
  }
  __syncthreads();

  for (int p = tid; p < NPIX; p += 256) {
    int y = p >> 5, x = p & 31;
    float s = 0.f;
    #pragma unroll
    for (int dy = 0; dy < 3; dy++)
      #pragma unroll
      for (int dx = 0; dx < 3; dx++) s += ed[y + dy][x + dx];
    size_t o = (size_t)b * HW + (size_t)(y0 + y) * WW + (x0 + x);
    edge[o] = ed[y + 1][x + 1];
    norm[o] = s * (1.f / 9.f) + 1e-6f;
  }
}

// ---------------------------------------------------------------------------
// Kernel 2: Q0 = softmax(unary) over the class dim
// ---------------------------------------------------------------------------
__global__ __launch_bounds__(256) void crf_softmax_init(
    const float* __restrict__ unary, float* __restrict__ Q)
{
  const size_t HW = (size_t)HH * WW;
  size_t p = (size_t)blockIdx.x * blockDim.x + threadIdx.x;   // over B*H*W
  if (p >= (size_t)BB * HW) return;
  size_t b = p / HW, off = p % HW;
  const float* u = unary + (size_t)b * CNUM * HW + off;
  float v[CNUM];
  float mx = -3.0e38f;
  #pragma unroll
  for (int c = 0; c < CNUM; c++) { v[c] = u[(size_t)c * HW]; mx = fmaxf(mx, v[c]); }
  float s = 0.f;
  #pragma unroll
  for (int c = 0; c < CNUM; c++) { v[c] = __expf(v[c] - mx); s += v[c]; }
  float inv = 1.f / s;
  float* q = Q + (size_t)b * CNUM * HW + off;
  #pragma unroll
  for (int c = 0; c < CNUM; c++) q[(size_t)c * HW] = v[c] * inv;
}

// ---------------------------------------------------------------------------
// Kernel 3: one fused mean-field iteration.
//   messages = sw*avgpool5(Q) + bw*avgpool3(Q*edge)/norm   (separable, LDS)
//   compat   = compatibility @ messages                    (WMMA f32 16x16x4)
//   Qout     = softmax(unary - compat)
// ---------------------------------------------------------------------------
__global__ __launch_bounds__(256) void crf_iter(
    const float* __restrict__ Qin, float* __restrict__ Qout,
    const float* __restrict__ unary, const float* __restrict__ edge,
    const float* __restrict__ norm, const float* __restrict__ compat,
    const float* __restrict__ swp, const float* __restrict__ bwp)
{
  __shared__ float et[18][35];          // edge, halo 1
  __shared__ float nt[16][32];          // normalizer
  __shared__ float qt[20][37];          // Q channel tile, halo 2
  __shared__ float qe[18][35];          // Q*edge, halo 1
  __shared__ float h5[20][33];          // horizontal 5-sum
  __shared__ float h3[18][33];          // horizontal 3-sum
  __shared__ float scr[CNUM][SCR_STRIDE]; // messages -> (in place) compat result

  const int tid  = threadIdx.x;
  const int lane = tid & 31;
  const int wave = tid >> 5;
  const int x0 = blockIdx.x * TW, y0 = blockIdx.y * TH, b = blockIdx.z;
  const size_t HW = (size_t)HH * WW;

  const float sw = fmaxf(swp[0], 0.f);
  const float bw = fmaxf(bwp[0], 0.f);

  // A-tiles for WMMA: A[m][k] = compat[m][k], zero-padded to 32x24.
  // 16x4 f32 A layout: lanes 0-15 hold row M=lane, K={k0,k0+1}; lanes 16-31 K={k0+2,k0+3}.
  v2f A[2][6];
  {
    int row0  = lane & 15;
    int khalf = (lane >> 4) * 2;
    #pragma unroll
    for (int t = 0; t < 2; t++) {
      int row = row0 + 16 * t;
      #pragma unroll
      for (int ks = 0; ks < 6; ks++) {
        int k = ks * 4 + khalf;
        v2f a;
        a.x = (row < CNUM && k     < CNUM) ? compat[row * CNUM + k]     : 0.f;
        a.y = (row < CNUM && k + 1 < CNUM) ? compat[row * CNUM + k + 1] : 0.f;
        A[t][ks] = a;
      }
    }
  }

  // Stage edge (halo 1) and norm tiles.
  for (int idx = tid; idx < 18 * 34; idx += 256) {
    int j = idx / 34, i = idx % 34;
    int ay = y0 - 1 + j, ax = x0 - 1 + i;
    et[j][i] = (ay >= 0 && ay < HH && ax >= 0 && ax < WW)
                 ? edge[(size_t)b * HW + (size_t)ay * WW + ax] : 0.f;
  }
  for (int p = tid; p < NPIX; p += 256) {
    int y = p >> 5, x = p & 31;
    nt[y][x] = norm[(size_t)b * HW + (size_t)(y0 + y) * WW + (x0 + x)];
  }
  __syncthreads();

  // ---- stencil stage: per-channel separable pools -> messages in scr ----
  for (int c = 0; c < CNUM; c++) {
    const float* qsrc = Qin + ((size_t)b * CNUM + c) * HW;
    for (int idx = tid; idx < 20 * 36; idx += 256) {
      int j = idx / 36, i = idx % 36;
      int ay = y0 - 2 + j, ax = x0 - 2 + i;
      qt[j][i] = (ay >= 0 && ay < HH && ax >= 0 && ax < WW)
                   ? qsrc[(size_t)ay * WW + ax] : 0.f;
    }
    __syncthreads();
    for (int idx = tid; idx < 18 * 34; idx += 256) {
      int j = idx / 34, i = idx % 34;
      qe[j][i] = qt[j + 1][i + 1] * et[j][i];
    }
    for (int idx = tid; idx < 20 * 32; idx += 256) {
      int j = idx >> 5, i = idx & 31;
      h5[j][i] = qt[j][i] + qt[j][i+1] + qt[j][i+2] + qt[j][i+3] + qt[j][i+4];
    }
    __syncthreads();
    for (int idx = tid; idx < 18 * 32; idx += 256) {
      int j = idx >> 5, i = idx & 31;
      h3[j][i] = qe[j][i] + qe[j][i+1] + qe[j][i+2];
    }
    __syncthreads();
    for (int p = tid; p < NPIX; p += 256) {
      int y = p >> 5, x = p & 31;
      float s5 = h5[y][x] + h5[y+1][x] + h5[y+2][x] + h5[y+3][x] + h5[y+4][x];
      float s3 = h3[y][x] + h3[y+1][x] + h3[y+2][x];
      scr[c][p] = sw * s5 * (1.f / 25.f) + bw * (s3 * (1.f / 9.f)) / nt[y][x];
    }
    __syncthreads();
  }

  // ---- WMMA compat transform: wave w owns pixel groups [4w, 4w+4) ----
  // D = compat(32x24 padded) x messages(24x16 padded) per 16-pixel group.
  {
    int khalf = (lane >> 4) * 2;
    int nidx  = lane & 15;
    for (int g4 = 0; g4 < 4; g4++) {
      int p0 = (wave * 4 + g4) * 16;
      v8f d0 = {}; v8f d1 = {};
      #pragma unroll
      for (int ks = 0; ks < 6; ks++) {
        int k = ks * 4 + khalf;
        v2f bv;
        bv.x = (k     < CNUM) ? scr[k][p0 + nidx]     : 0.f;
        bv.y = (k + 1 < CNUM) ? scr[k + 1][p0 + nidx] : 0.f;
        d0 = __builtin_amdgcn_wmma_f32_16x16x4_f32(false, A[0][ks], false, bv,
                                                   (short)0, d0, false, false);
        d1 = __builtin_amdgcn_wmma_f32_16x16x4_f32(false, A[1][ks], false, bv,
                                                   (short)0, d1, false, false);
      }
      // D layout: VGPR r -> M = r (+8 for lanes 16-31), N = lane&15.
      int mofs = (lane >= 16) ? 8 : 0;
      #pragma unroll
      for (int r = 0; r < 8; r++) scr[r + mofs][p0 + nidx] = d0[r];
      if (lane < 16) {
        #pragma unroll
        for (int r = 0; r < 5; r++) scr[16 + r][p0 + nidx] = d1[r];  // classes 16..20
      }
    }
  }
  __syncthreads();

  // ---- fused softmax(unary - compat) ----
  for (int p = tid; p < NPIX; p += 256) {
    int y = p >> 5, x = p & 31;
    size_t base = (size_t)b * CNUM * HW + (size_t)(y0 + y) * WW + (x0 + x);
    float v[CNUM];
    float mx = -3.0e38f;
    #pragma unroll
    for (int c = 0; c < CNUM; c++) {
      float l = unary[base + (size_t)c * HW] - scr[c][p];
      v[c] = l; mx = fmaxf(mx, l);
    }
    float s = 0.f;
    #pragma unroll
    for (int c = 0; c < CNUM; c++) { float e = __expf(v[c] - mx); v[c] = e; s += e; }
    float inv = 1.f / s;
    #pragma unroll
    for (int c = 0; c < CNUM; c++) Qout[base + (size_t)c * HW] = v[c] * inv;
  }
}

// ---------------------------------------------------------------------------
extern "C" void kernel_launch(void* const* d_in, const int* in_sizes, int n_in,
                              void* d_out, int out_size, void* d_ws, size_t ws_size,
                              hipStream_t stream)
{
  const float* unary  = (const float*)d_in[0];
  const float* image  = (const float*)d_in[1];
  const float* compat = (const float*)d_in[2];
  const float* swp    = (const float*)d_in[3];
  const float* bwp    = (const float*)d_in[4];

  const size_t HW = (size_t)HH * WW;
  char*  ws   = (char*)d_ws;
  float* edge = (float*)ws;                                   // B*H*W floats (8 MB)
  float* norm = (float*)(ws + (size_t)BB * HW * 4);           // 8 MB
  float* Qb   = (float*)(ws + (size_t)2 * BB * HW * 4);       // ping buffer (176 MB)
  float* Qa   = (float*)d_out;                                // final result lives here

  dim3 tgrid(WW / TW, HH / TH, BB);   // 16 x 32 x 8 tiles
  crf_edge_norm<<<tgrid, 256, 0, stream>>>(image, edge, norm);

  {
    size_t npix = (size_t)BB * HW;
    int blocks = (int)((npix + 255) / 256);
    crf_softmax_init<<<blocks, 256, 0, stream>>>(unary, Qa);
  }

  float* qin = Qa; float* qout = Qb;
  for (int it = 0; it < 10; it++) {       // even count: final write lands in Qa == d_out
    crf_iter<<<tgrid, 256, 0, stream>>>(qin, qout, unary, edge, norm, compat, swp, bwp);
    float* t = qin; qin = qout; qout = t;
  }
}